// Model_20968030339505
// MI455X (gfx1250) — compile-verified
//
#include <hip/hip_runtime.h>

#define NN 100000   // num nodes (users == items)
#define HH 128      // hidden channels
#define EE 1600000  // edges per edge type
#define LL 100000   // labeled edges

typedef float v2f __attribute__((ext_vector_type(2)));
typedef float v8f __attribute__((ext_vector_type(8)));

// ---------------- zero ----------------
__global__ __launch_bounds__(256) void zero_kernel(float* __restrict__ p, long n) {
  long i = (long)blockIdx.x * blockDim.x + threadIdx.x;
  long stride = (long)gridDim.x * blockDim.x;
  for (; i < n; i += stride) p[i] = 0.0f;
}

// ---------------- edge scatter-add (one wave per edge) ----------------
__global__ __launch_bounds__(256) void scatter_kernel(
    const float* __restrict__ xsrc, const int* __restrict__ srci,
    const int* __restrict__ dsti, float* __restrict__ agg,
    float* __restrict__ cnt, int E) {
  int gw = (int)(((unsigned)blockIdx.x * blockDim.x + threadIdx.x) >> 5);
  int lane = threadIdx.x & 31;
  if (gw >= E) return;
  int s = srci[gw], d = dsti[gw];
  const float4* sp = (const float4*)(xsrc + (size_t)s * HH);
  float4 v = sp[lane];
  float* ap = agg + (size_t)d * HH + lane * 4;
  atomicAdd(ap + 0, v.x);
  atomicAdd(ap + 1, v.y);
  atomicAdd(ap + 2, v.z);
  atomicAdd(ap + 3, v.w);
  if (lane == 0) atomicAdd(cnt + d, 1.0f);
}

// ------- SAGEConv GEMM: out[16 x 128] = mean@Wl + bias + xdst@Wr -------
// block = 256 threads = 8 waves; blockIdx.x = 16-row node tile; wave = 16-col tile.
// V_WMMA_F32_16X16X4_F32: A 16x4 (lane=M, vgprs={K,K+1}, upper half-wave K+=2),
// B 4x16 / C 16x16 striped across lanes (upper half-wave = rows +8 / K+2).
__global__ __launch_bounds__(256) void sage_gemm_kernel(
    const float* __restrict__ agg, const float* __restrict__ cnt,
    const float* __restrict__ xdst,
    const float* __restrict__ wl, const float* __restrict__ wr,
    const float* __restrict__ bias, float* __restrict__ out) {
  const int wv   = threadIdx.x >> 5;
  const int lane = threadIdx.x & 31;
  const int half = lane >> 4;
  const int l16  = lane & 15;
  const int row  = blockIdx.x * 16 + l16;   // A-row owned by this lane
  const int col  = wv * 16 + l16;           // B/C column owned by this lane

  const float rs = 1.0f / fmaxf(cnt[row], 1.0f);   // mean = sum / max(cnt,1)
  const float* A = agg  + (size_t)row * HH;
  const float* X = xdst + (size_t)row * HH;

  v8f c = {};
  // mean @ Wl
  #pragma unroll 4
  for (int k0 = 0; k0 < HH; k0 += 4) {
    const int ka = k0 + half * 2;
    v2f a, b;
    a[0] = A[ka] * rs;          a[1] = A[ka + 1] * rs;
    b[0] = wl[ka * HH + col];   b[1] = wl[(ka + 1) * HH + col];
    c = __builtin_amdgcn_wmma_f32_16x16x4_f32(false, a, false, b, (short)0, c,
                                              false, false);
  }
  // x_dst @ Wr
  #pragma unroll 4
  for (int k0 = 0; k0 < HH; k0 += 4) {
    const int ka = k0 + half * 2;
    v2f a, b;
    a[0] = X[ka];               a[1] = X[ka + 1];
    b[0] = wr[ka * HH + col];   b[1] = wr[(ka + 1) * HH + col];
    c = __builtin_amdgcn_wmma_f32_16x16x4_f32(false, a, false, b, (short)0, c,
                                              false, false);
  }
  // C layout: vgpr r -> M = r + half*8, N = col
  const float bv = bias[col];
  const int rbase = blockIdx.x * 16 + half * 8;
  #pragma unroll
  for (int r = 0; r < 8; ++r)
    out[(size_t)(rbase + r) * HH + col] = c[r] + bv;
}

// ---------------- per-row L2-normalize + ReLU, in place ----------------
__global__ __launch_bounds__(256) void norm_relu_kernel(float* __restrict__ x, int N) {
  int rowq = (int)(((unsigned)blockIdx.x * blockDim.x + threadIdx.x) >> 5);
  int lane = threadIdx.x & 31;
  if (rowq >= N) return;
  float4* p = (float4*)(x + (size_t)rowq * HH);
  float4 v = p[lane];
  float ss = v.x * v.x + v.y * v.y + v.z * v.z + v.w * v.w;
  #pragma unroll
  for (int off = 16; off > 0; off >>= 1) ss += __shfl_xor(ss, off, 32);
  float inv = 1.0f / fmaxf(sqrtf(ss), 1e-12f);
  v.x = fmaxf(v.x * inv, 0.0f);
  v.y = fmaxf(v.y * inv, 0.0f);
  v.z = fmaxf(v.z * inv, 0.0f);
  v.w = fmaxf(v.w * inv, 0.0f);
  p[lane] = v;
}

// ---------------- classifier: dot(user[src], item[dst]) ----------------
__global__ __launch_bounds__(256) void dot_kernel(
    const float* __restrict__ xu, const float* __restrict__ xi,
    const int* __restrict__ su, const int* __restrict__ di,
    float* __restrict__ out, int L) {
  int e = (int)(((unsigned)blockIdx.x * blockDim.x + threadIdx.x) >> 5);
  int lane = threadIdx.x & 31;
  if (e >= L) return;
  const float4* up = (const float4*)(xu + (size_t)su[e] * HH);
  const float4* ip = (const float4*)(xi + (size_t)di[e] * HH);
  float4 a = up[lane], b = ip[lane];
  float s = a.x * b.x + a.y * b.y + a.z * b.z + a.w * b.w;
  #pragma unroll
  for (int off = 16; off > 0; off >>= 1) s += __shfl_xor(s, off, 32);
  if (lane == 0) out[e] = s;
}

extern "C" void kernel_launch(void* const* d_in, const int* in_sizes, int n_in,
                              void* d_out, int out_size, void* d_ws, size_t ws_size,
                              hipStream_t stream) {
  const float* emb_user  = (const float*)d_in[0];
  const float* emb_item  = (const float*)d_in[1];
  const int*   e_rates   = (const int*)d_in[2];   // [2,E] row0=src(user) row1=dst(item)
  const int*   e_rev     = (const int*)d_in[3];
  const int*   e_lab     = (const int*)d_in[4];   // [2,L]
  const float* w1l_rates = (const float*)d_in[5];
  const float* w1r_rates = (const float*)d_in[6];
  const float* w1l_rev   = (const float*)d_in[7];
  const float* w1r_rev   = (const float*)d_in[8];
  const float* w2l_rates = (const float*)d_in[9];
  const float* w2r_rates = (const float*)d_in[10];
  const float* w2l_rev   = (const float*)d_in[11];
  const float* w2r_rev   = (const float*)d_in[12];
  const float* b1_rates  = (const float*)d_in[13];
  const float* b1_rev    = (const float*)d_in[14];
  const float* b2_rates  = (const float*)d_in[15];
  const float* b2_rev    = (const float*)d_in[16];
  float* out = (float*)d_out;

  // workspace layout (fp32): agg | cnt | xi1 | xu1 | xi2 | xu2  (~256 MB)
  float* agg = (float*)d_ws;
  float* cnt = agg + (size_t)NN * HH;
  float* xi1 = cnt + NN;
  float* xu1 = xi1 + (size_t)NN * HH;
  float* xi2 = xu1 + (size_t)NN * HH;
  float* xu2 = xi2 + (size_t)NN * HH;

  dim3 b256(256);
  const int zgrid = 2048;
  const int sgrid = EE / 8;        // 8 edges (waves) per block
  const int ggrid = NN / 16;       // 6250 node tiles (exact)
  const int ngrid = (NN + 7) / 8;  // 8 rows per block
  const int dgrid = (LL + 7) / 8;

  auto conv = [&](const float* xsrc, const float* xdst, const int* edges,
                  const float* wl, const float* wr, const float* bias,
                  float* xout) {
    zero_kernel<<<zgrid, b256, 0, stream>>>(agg, (long)NN * HH);
    zero_kernel<<<zgrid, b256, 0, stream>>>(cnt, (long)NN);
    scatter_kernel<<<sgrid, b256, 0, stream>>>(xsrc, edges, edges + EE, agg, cnt, EE);
    sage_gemm_kernel<<<ggrid, b256, 0, stream>>>(agg, cnt, xdst, wl, wr, bias, xout);
    norm_relu_kernel<<<ngrid, b256, 0, stream>>>(xout, NN);
  };

  // layer 1
  conv(emb_user, emb_item, e_rates, w1l_rates, w1r_rates, b1_rates, xi1);
  conv(emb_item, emb_user, e_rev,   w1l_rev,   w1r_rev,   b1_rev,   xu1);
  // layer 2
  conv(xu1, xi1, e_rates, w2l_rates, w2r_rates, b2_rates, xi2);
  conv(xi1, xu1, e_rev,   w2l_rev,   w2r_rev,   b2_rev,   xu2);
  // classifier
  dot_kernel<<<dgrid, b256, 0, stream>>>(xu2, xi2, e_lab, e_lab + LL, out, LL);
}